// GCNLayer_61589831025106
// MI455X (gfx1250) — compile-verified
//
#include <hip/hip_runtime.h>

typedef float v2f __attribute__((ext_vector_type(2)));
typedef float v8f __attribute__((ext_vector_type(8)));

#define IN_DIM 128
#define OUT_DIM 128

// ---------------------------------------------------------------------------
// Kernel 1: hw[n, :] = (h[n, :] @ W) * norm[n]   via V_WMMA_F32_16X16X4_F32
// One wave per 16-row tile; the wave computes all 8 column tiles (128 cols).
// A fragment (16x4 f32, 2 VGPRs) loaded once per k-step, reused by 8 WMMAs.
// ---------------------------------------------------------------------------
__global__ __launch_bounds__(256) void gcn_gemm_norm(
    const float* __restrict__ h, const float* __restrict__ W,
    const float* __restrict__ norm, float* __restrict__ hw,
    int n_nodes, int n_row_tiles)
{
    const int lane = threadIdx.x & 31;
    const int wave = threadIdx.x >> 5;                 // 0..7
    const int tile = blockIdx.x * 8 + wave;            // row tile index
    if (tile >= n_row_tiles) return;                   // wave-uniform exit

    const int m0 = tile * 16;
    const int lm = lane & 15;                          // 0..15
    const int lh = lane >> 4;                          // 0 or 1

    // A-fragment row for this lane (clamped so OOB never loads past end)
    int arow = m0 + lm;
    if (arow >= n_nodes) arow = n_nodes - 1;
    const float* __restrict__ hrow = h + (long long)arow * IN_DIM;

    v8f acc[8];
    #pragma unroll
    for (int t = 0; t < 8; ++t) acc[t] = (v8f){};

    for (int k = 0; k < IN_DIM; k += 4) {
        // A: 16x4 f32. lanes 0-15: K=k,k+1  lanes 16-31: K=k+2,k+3
        const int ka = k + 2 * lh;
        v2f a;
        a.x = hrow[ka];
        a.y = hrow[ka + 1];

        // B: 4x16 f32 per column tile. lanes 0-15: K=ka row, N=lm
        #pragma unroll
        for (int t = 0; t < 8; ++t) {
            const int n0 = t * 16;
            v2f b;
            b.x = W[(long long)ka * OUT_DIM + n0 + lm];
            b.y = W[(long long)(ka + 1) * OUT_DIM + n0 + lm];
            acc[t] = __builtin_amdgcn_wmma_f32_16x16x4_f32(
                false, a, false, b, (short)0, acc[t], false, false);
        }
    }

    // D layout (16x16 f32, 8 VGPRs): VGPR r -> M = r + 8*lh, N = lm
    #pragma unroll
    for (int r = 0; r < 8; ++r) {
        const int m = m0 + r + 8 * lh;
        if (m < n_nodes) {
            const float nv = norm[m];
            #pragma unroll
            for (int t = 0; t < 8; ++t) {
                hw[(long long)m * OUT_DIM + t * 16 + lm] = acc[t][r] * nv;
            }
        }
    }
}

// ---------------------------------------------------------------------------
// Kernel 2: zero the accumulation buffer (d_out)
// ---------------------------------------------------------------------------
__global__ __launch_bounds__(256) void gcn_zero(float4* __restrict__ p, long long n4)
{
    long long gid = (long long)blockIdx.x * blockDim.x + threadIdx.x;
    if (gid < n4) p[gid] = make_float4(0.f, 0.f, 0.f, 0.f);
}

// ---------------------------------------------------------------------------
// Kernel 3: segment-sum via f32 atomics. One wave per edge; each lane moves
// 16B (float4 load, 4 atomic adds). Wave reads the full 512B source row
// coalesced.
// ---------------------------------------------------------------------------
__global__ __launch_bounds__(256) void gcn_scatter(
    const float* __restrict__ hw, const int* __restrict__ src,
    const int* __restrict__ dst, float* __restrict__ out, long long total_thr)
{
    long long gid = (long long)blockIdx.x * blockDim.x + threadIdx.x;
    if (gid >= total_thr) return;
    const int e  = (int)(gid >> 5);
    const int f4 = (int)(gid & 31) * 4;

    const int s = src[e];
    const int d = dst[e];

    const float4 v = *reinterpret_cast<const float4*>(
        hw + (long long)s * OUT_DIM + f4);
    float* o = out + (long long)d * OUT_DIM + f4;
    atomicAdd(o + 0, v.x);
    atomicAdd(o + 1, v.y);
    atomicAdd(o + 2, v.z);
    atomicAdd(o + 3, v.w);
}

// ---------------------------------------------------------------------------
// Kernel 4: out = out * norm[node] + b[feat]   (in place, float4)
// ---------------------------------------------------------------------------
__global__ __launch_bounds__(256) void gcn_finalize(
    float4* __restrict__ out, const float* __restrict__ norm,
    const float4* __restrict__ bias4, long long n4)
{
    long long gid = (long long)blockIdx.x * blockDim.x + threadIdx.x;
    if (gid >= n4) return;
    const int node = (int)(gid >> 5);        // 32 float4 per node row
    const int f4   = (int)(gid & 31);
    const float nv = norm[node];
    float4 v = out[gid];
    const float4 b = bias4[f4];
    v.x = v.x * nv + b.x;
    v.y = v.y * nv + b.y;
    v.z = v.z * nv + b.z;
    v.w = v.w * nv + b.w;
    out[gid] = v;
}

// ---------------------------------------------------------------------------
extern "C" void kernel_launch(void* const* d_in, const int* in_sizes, int n_in,
                              void* d_out, int out_size, void* d_ws, size_t ws_size,
                              hipStream_t stream)
{
    const float* h    = (const float*)d_in[0];
    const float* norm = (const float*)d_in[1];
    const float* W    = (const float*)d_in[2];
    const float* bias = (const float*)d_in[3];
    const int*   src  = (const int*)d_in[4];
    const int*   dst  = (const int*)d_in[5];
    float* out = (float*)d_out;
    float* hw  = (float*)d_ws;                 // N x 128 f32 scratch (25.6 MB)

    const int n_nodes = in_sizes[1];           // norm has N elements
    const int n_edges = in_sizes[4];

    // 1) projection + pre-norm -> hw
    const int n_row_tiles = (n_nodes + 15) / 16;
    const int gemm_blocks = (n_row_tiles + 7) / 8;       // 8 waves / block
    gcn_gemm_norm<<<gemm_blocks, 256, 0, stream>>>(h, W, norm, hw,
                                                   n_nodes, n_row_tiles);

    // 2) zero accumulator (d_out)
    const long long n4 = (long long)n_nodes * (OUT_DIM / 4);
    const int zb = (int)((n4 + 255) / 256);
    gcn_zero<<<zb, 256, 0, stream>>>((float4*)out, n4);

    // 3) edge scatter-add
    const long long sc_thr = (long long)n_edges * 32;
    const int sb = (int)((sc_thr + 255) / 256);
    gcn_scatter<<<sb, 256, 0, stream>>>(hw, src, dst, out, sc_thr);

    // 4) post-norm + bias
    gcn_finalize<<<zb, 256, 0, stream>>>((float4*)out, norm,
                                         (const float4*)bias, n4);
}